// Sage_45423574122804
// MI455X (gfx1250) — compile-verified
//
#include <hip/hip_runtime.h>

typedef __attribute__((ext_vector_type(2))) float v2f;
typedef __attribute__((ext_vector_type(8))) float v8f;

// ---------------------------------------------------------------------------
// Zero a float region (grid-stride).
// ---------------------------------------------------------------------------
__global__ void zero_kernel(float* __restrict__ p, long nf) {
    long i = (long)blockIdx.x * blockDim.x + threadIdx.x;
    long stride = (long)gridDim.x * blockDim.x;
    for (; i < nf; i += stride) p[i] = 0.0f;
}

// ---------------------------------------------------------------------------
// In-degree via float atomics: deg[dst[e]] += 1
// (reads only the low dword of the int64 index; ids < 2^31)
// ---------------------------------------------------------------------------
__global__ void deg_kernel(const int* __restrict__ dst32,
                           float* __restrict__ deg, int n_edges) {
    int e = blockIdx.x * blockDim.x + threadIdx.x;
    if (e < n_edges) atomicAdd(&deg[dst32[(size_t)e * 2]], 1.0f);
}

// deg -> 1/max(deg,1) in place
__global__ void inv_kernel(float* __restrict__ deg, int n) {
    int i = blockIdx.x * blockDim.x + threadIdx.x;
    if (i < n) deg[i] = 1.0f / fmaxf(deg[i], 1.0f);
}

// ---------------------------------------------------------------------------
// Edge-parallel scatter-add: agg[dst] += feat[src].
// Each thread handles 4 features of one edge (float4 gather, 4 atomics).
// LG_GROUPS = log2(DIM/4).
// ---------------------------------------------------------------------------
template <int DIM, int LG_GROUPS>
__global__ void scatter_kernel(const float* __restrict__ feat,
                               const int* __restrict__ src32,
                               const int* __restrict__ dst32,
                               float* __restrict__ agg, int n_edges) {
    long t = (long)blockIdx.x * blockDim.x + threadIdx.x;
    long e = t >> LG_GROUPS;
    int  g = (int)(t & ((1 << LG_GROUPS) - 1));
    if (e >= n_edges) return;
    int s = src32[e * 2];
    int d = dst32[e * 2];
    const float4 v = *(const float4*)(feat + (size_t)s * DIM + g * 4);
    float* base = agg + (size_t)d * DIM + g * 4;
    atomicAdd(base + 0, v.x);
    atomicAdd(base + 1, v.y);
    atomicAdd(base + 2, v.z);
    atomicAdd(base + 3, v.w);
}

// ---------------------------------------------------------------------------
// Fused SAGE dense stage with V_WMMA_F32_16X16X4_F32 (fp32 precision):
//   out = act( scale(Aagg) @ Wl  +  Aself @ Wr  +  bias )
//
// * Weights staged in LDS, zero-padded to WPAD = 16*NCB columns, stored as
//   K-pair interleaved v2f so each B fragment is ONE unguarded ds_load_b64.
// * One wave per 16-node tile; each A fragment is shared by all NCB column
//   blocks (accumulators live simultaneously).
// * All trip counts compile-time -> fully unrolled WMMA chains.
//
// Fragment layouts per CDNA5 ISA 7.12.2:
//   A 16x4 f32 : lanes 0-15 -> {K=0,1}, lanes 16-31 -> {K=2,3}, M=lane&15
//   B 4x16  f32: lane holds column (lane&15); same K split as A
//   C/D 16x16  : VGPR r -> row r / r+8 (lane half), N = lane&15
// ---------------------------------------------------------------------------
template <int K, int NOUT, bool HAS_SELF, bool HAS_BIAS, bool RELU, bool HAS_SCALE>
__global__ void sage_gemm_wmma(const float* __restrict__ Aagg,     // [n,K]
                               const float* __restrict__ inv_deg,  // [n] (if HAS_SCALE)
                               const float* __restrict__ Aself,    // [n,K] (if HAS_SELF)
                               const float* __restrict__ Wl,       // [K,NOUT]
                               const float* __restrict__ Wr,       // [K,NOUT] (if HAS_SELF)
                               const float* __restrict__ bias,     // [NOUT] (if HAS_BIAS)
                               float* __restrict__ out,            // [n,NOUT]
                               int n_nodes) {
    constexpr int NCB  = (NOUT + 15) / 16;  // 16-wide output column blocks
    constexpr int WPAD = NCB * 16;          // padded width
    constexpr int KK   = K / 4;             // WMMA K-steps

    // LDS: weight pair (rows 2p, 2p+1) for column n at [p*WPAD + n]
    __shared__ v2f   lds_wl[(K / 2) * WPAD];
    __shared__ v2f   lds_wr[HAS_SELF ? (K / 2) * WPAD : 1];
    __shared__ float lds_b[WPAD];

    for (int i = threadIdx.x; i < (K / 2) * WPAD; i += blockDim.x) {
        const int p = i / WPAD;       // K pair index (compile-time divisor)
        const int n = i - p * WPAD;   // padded column
        const bool ok = (n < NOUT);
        v2f wv;
        wv.x = ok ? Wl[(2 * p) * NOUT + n] : 0.0f;
        wv.y = ok ? Wl[(2 * p + 1) * NOUT + n] : 0.0f;
        lds_wl[i] = wv;
        if (HAS_SELF) {
            v2f sv;
            sv.x = ok ? Wr[(2 * p) * NOUT + n] : 0.0f;
            sv.y = ok ? Wr[(2 * p + 1) * NOUT + n] : 0.0f;
            lds_wr[i] = sv;
        }
    }
    if (threadIdx.x < WPAD)
        lds_b[threadIdx.x] =
            (HAS_BIAS && threadIdx.x < NOUT) ? bias[threadIdx.x] : 0.0f;
    __syncthreads();

    const int lane = threadIdx.x & 31;
    const int wave = threadIdx.x >> 5;
    const int tile = blockIdx.x * (blockDim.x >> 5) + wave;
    const int base = tile * 16;
    if (base >= n_nodes) return;  // wave-uniform: EXEC all-ones at WMMA

    const int half = lane >> 4;   // 0 -> K pair {0,1}; 1 -> K pair {2,3}
    const int mrow = lane & 15;
    int node = base + mrow;
    if (node >= n_nodes) node = n_nodes - 1;  // clamp (N%16==0 in practice)
    const float scale = HAS_SCALE ? inv_deg[node] : 1.0f;

    v8f acc[NCB];
#pragma unroll
    for (int c = 0; c < NCB; ++c)
        acc[c] = (v8f){0.f, 0.f, 0.f, 0.f, 0.f, 0.f, 0.f, 0.f};

    // ---- mean-aggregated path: scale(Aagg) @ Wl ----
    {
        const float* Arow = Aagg + (size_t)node * K;
#pragma unroll
        for (int kk = 0; kk < KK; ++kk) {
            const int k0 = kk * 4 + half * 2;        // even -> 8B aligned
            v2f a = *(const v2f*)(Arow + k0);
            if (HAS_SCALE) { a.x *= scale; a.y *= scale; }
            const int prow = kk * 2 + half;          // K pair row in LDS
#pragma unroll
            for (int c = 0; c < NCB; ++c) {
                v2f b = lds_wl[prow * WPAD + mrow + c * 16];
                acc[c] = __builtin_amdgcn_wmma_f32_16x16x4_f32(
                    false, a, false, b, (short)0, acc[c], false, false);
            }
        }
    }
    // ---- self path: Aself @ Wr ----
    if (HAS_SELF) {
        const float* Srow = Aself + (size_t)node * K;
#pragma unroll
        for (int kk = 0; kk < KK; ++kk) {
            const int k0 = kk * 4 + half * 2;
            v2f a = *(const v2f*)(Srow + k0);
            const int prow = kk * 2 + half;
#pragma unroll
            for (int c = 0; c < NCB; ++c) {
                v2f b = lds_wr[prow * WPAD + mrow + c * 16];
                acc[c] = __builtin_amdgcn_wmma_f32_16x16x4_f32(
                    false, a, false, b, (short)0, acc[c], false, false);
            }
        }
    }

    // ---- epilogue: bias + relu + store (store mask only where NOUT%16) ----
#pragma unroll
    for (int c = 0; c < NCB; ++c) {
        const int ncol = mrow + c * 16;
        const float bv = lds_b[ncol];
#pragma unroll
        for (int r = 0; r < 8; ++r) {
            const int onode = base + r + half * 8;
            float v = acc[c][r] + bv;
            if (RELU) v = fmaxf(v, 0.0f);
            bool store_ok = (onode < n_nodes);
            if (NOUT % 16 != 0 && c == NCB - 1) store_ok = store_ok && (ncol < NOUT);
            if (store_ok) out[(size_t)onode * NOUT + ncol] = v;
        }
    }
}

// ---------------------------------------------------------------------------
// Driver.  Inputs (setup_inputs order):
//   0:x[N,64] f32  1:edge_index[2,E] i64  2:W1l[64,32]  3:b1[32]  4:W1r[64,32]
//   5:W2l[32,32]   6:b2[32]  7:W2r[32,32] 8:w[32,40]
// Output: [N,40] f32
// ---------------------------------------------------------------------------
extern "C" void kernel_launch(void* const* d_in, const int* in_sizes, int n_in,
                              void* d_out, int out_size, void* d_ws, size_t ws_size,
                              hipStream_t stream) {
    const float* x   = (const float*)d_in[0];
    const int*   ei  = (const int*)d_in[1];   // int64 viewed as dword pairs
    const float* W1l = (const float*)d_in[2];
    const float* b1  = (const float*)d_in[3];
    const float* W1r = (const float*)d_in[4];
    const float* W2l = (const float*)d_in[5];
    const float* b2  = (const float*)d_in[6];
    const float* W2r = (const float*)d_in[7];
    const float* w   = (const float*)d_in[8];

    const int IN_DIM = 64, HID = 32;
    const int n = in_sizes[0] / IN_DIM;
    const int E = in_sizes[1] / 2;
    const int* src32 = ei;          // low dwords of edge_index[0]
    const int* dst32 = ei + 2 * E;  // low dwords of edge_index[1]

    // Workspace layout: deg | agg1 | agg2 | h1 | h2
    float* ws   = (float*)d_ws;
    float* deg  = ws;
    float* agg1 = deg  + (size_t)n;
    float* agg2 = agg1 + (size_t)n * IN_DIM;
    float* h1   = agg2 + (size_t)n * HID;
    float* h2   = h1   + (size_t)n * HID;

    const int T = 256;

    // 1. Zero deg + agg1 + agg2 (contiguous) every call.
    const long nz = (long)n * (1 + IN_DIM + HID);
    zero_kernel<<<2048, T, 0, stream>>>(ws, nz);

    // 2. In-degree, then 1/max(deg,1) in place.
    deg_kernel<<<(E + T - 1) / T, T, 0, stream>>>(dst32, deg, E);
    inv_kernel<<<(n + T - 1) / T, T, 0, stream>>>(deg, n);

    // 3. Layer-1 scatter: agg1[dst] += x[src]   (16 x float4 per edge)
    {
        long tot = (long)E << 4;
        scatter_kernel<64, 4><<<(int)((tot + T - 1) / T), T, 0, stream>>>(
            x, src32, dst32, agg1, E);
    }

    // 4. Layer-1 dense (WMMA): h1 = relu(mean(agg1)@W1l + b1 + x@W1r)
    const int tiles = (n + 15) / 16;
    const int wpb = 4;  // 4 waves / 128-thread block
    const int ggrid = (tiles + wpb - 1) / wpb;
    sage_gemm_wmma<64, 32, true, true, true, true>
        <<<ggrid, 32 * wpb, 0, stream>>>(agg1, deg, x, W1l, W1r, b1, h1, n);

    // 5. Layer-2 scatter: agg2[dst] += h1[src]  (8 x float4 per edge)
    {
        long tot = (long)E << 3;
        scatter_kernel<32, 3><<<(int)((tot + T - 1) / T), T, 0, stream>>>(
            h1, src32, dst32, agg2, E);
    }

    // 6. Layer-2 dense (WMMA): h2 = relu(mean(agg2)@W2l + b2 + h1@W2r)
    sage_gemm_wmma<32, 32, true, true, true, true>
        <<<ggrid, 32 * wpb, 0, stream>>>(agg2, deg, h1, W2l, W2r, b2, h2, n);

    // 7. Head (WMMA): out = h2 @ w   (NOUT=40 -> 3 col blocks, masked store)
    sage_gemm_wmma<32, 40, false, false, false, false>
        <<<ggrid, 32 * wpb, 0, stream>>>(h2, nullptr, nullptr, w, nullptr,
                                         nullptr, (float*)d_out, n);
}